// TopK_64484638982222
// MI455X (gfx1250) — compile-verified
//
#include <hip/hip_runtime.h>
#include <hip/hip_bf16.h>

typedef __attribute__((ext_vector_type(2))) float v2f;
typedef __attribute__((ext_vector_type(4))) float v4f;
typedef __attribute__((ext_vector_type(8))) float v8f;

#if defined(__AMDGCN__) && defined(__gfx1250__) && __has_builtin(__builtin_amdgcn_wmma_f32_16x16x4_f32)
#define USE_WMMA_SUM 1
#else
#define USE_WMMA_SUM 0
#endif

#define HIST_BINS 256
#define NBLOCKS   1024
#define NTHREADS  256
#define CACHE_OFF 8192   // byte offset of the loss cache inside d_ws

// ---------- helpers ----------

// stable BCE-with-logits w/ pos_weight, reduction='none'
// softplus(-x) = max(-x,0) + log(1 + exp(-|x|)); arg of log is in (1,2] -> no cancellation
__device__ __forceinline__ float bce_loss(float x, float t, float pw) {
    float z  = -x;
    float sp = fmaxf(z, 0.0f) + __logf(1.0f + __expf(-fabsf(z)));   // v_exp_f32 / v_log_f32
    return (1.0f - t) * x + (1.0f + (pw - 1.0f) * t) * sp;
}

// monotonic float -> uint key (order-preserving, handles negatives)
__device__ __forceinline__ unsigned f2k(float f) {
    unsigned u = __float_as_uint(f);
    return u ^ ((unsigned)((int)u >> 31) | 0x80000000u);
}
__device__ __forceinline__ float k2f(unsigned k) {
    unsigned u = (k & 0x80000000u) ? (k ^ 0x80000000u) : (k ^ 0xFFFFFFFFu);
    return __uint_as_float(u);
}

// state (uints): st[0]=prefix  st[1]=k_remaining  st[2]=count_strictly_greater  st[3]=threshold key

// ---------- kernels ----------

__global__ void tk_init(unsigned* __restrict__ hist, unsigned* __restrict__ st,
                        const int* __restrict__ kp, int n) {
    for (int i = 0; i < HIST_BINS; ++i) hist[i] = 0u;
    long long K = ((long long)n * (long long)kp[0]) / 100ll;
    st[0] = 0u; st[1] = (unsigned)K; st[2] = 0u; st[3] = 0u;
}

// Pass 0 + loss cache fill. pred/targ are streamed exactly once in the cached
// pipeline -> non-temporal loads (TH=NT) so the 128MB stream does not evict the
// 64MB loss cache, which fits in (192MB) L2 and feeds all later passes at L2 rate.
__global__ void tk_p0_cache(const float* __restrict__ pred, const float* __restrict__ targ,
                            const float* __restrict__ pwp, int n,
                            unsigned* __restrict__ hist, float* __restrict__ cache) {
    __shared__ unsigned lh[HIST_BINS];
    const int tid = threadIdx.x;
    for (int i = tid; i < HIST_BINS; i += blockDim.x) lh[i] = 0u;
    __syncthreads();

    const float pw  = pwp[0];
    const int   gid = blockIdx.x * blockDim.x + tid;
    const int   gsz = gridDim.x * blockDim.x;
    const int   n4  = n >> 2;

    const v4f* p4 = (const v4f*)pred;
    const v4f* t4 = (const v4f*)targ;
    v4f*       c4 = (v4f*)cache;

    for (int i = gid; i < n4; i += gsz) {
        v4f p = __builtin_nontemporal_load(p4 + i);   // global_load_b128 (NT)
        v4f t = __builtin_nontemporal_load(t4 + i);
        v4f l;
        l.x = bce_loss(p.x, t.x, pw);
        l.y = bce_loss(p.y, t.y, pw);
        l.z = bce_loss(p.z, t.z, pw);
        l.w = bce_loss(p.w, t.w, pw);
        c4[i] = l;                                    // RT store: stays L2-resident
        atomicAdd(&lh[f2k(l.x) >> 24], 1u);
        atomicAdd(&lh[f2k(l.y) >> 24], 1u);
        atomicAdd(&lh[f2k(l.z) >> 24], 1u);
        atomicAdd(&lh[f2k(l.w) >> 24], 1u);
    }
    for (int i = (n4 << 2) + gid; i < n; i += gsz) {  // tail (n % 4)
        float p = __builtin_nontemporal_load(pred + i);
        float t = __builtin_nontemporal_load(targ + i);
        float l = bce_loss(p, t, pw);
        cache[i] = l;
        atomicAdd(&lh[f2k(l) >> 24], 1u);
    }
    __syncthreads();
    for (int i = tid; i < HIST_BINS; i += blockDim.x)
        if (lh[i]) atomicAdd(&hist[i], lh[i]);
}

// Passes 1..3 from the cached loss values (L2-resident reads).
__global__ void tk_hist_cache(const float* __restrict__ cache, int n,
                              unsigned* __restrict__ hist, const unsigned* __restrict__ st,
                              int pass) {
    __shared__ unsigned lh[HIST_BINS];
    const int tid = threadIdx.x;
    for (int i = tid; i < HIST_BINS; i += blockDim.x) lh[i] = 0u;
    __syncthreads();

    const unsigned prefix = st[0];
    const int shift  = 24 - 8 * pass;
    const int hshift = 32 - 8 * pass;      // pass >= 1 here
    const int gid = blockIdx.x * blockDim.x + tid;
    const int gsz = gridDim.x * blockDim.x;
    const int n4  = n >> 2;
    const v4f* c4 = (const v4f*)cache;

    for (int i = gid; i < n4; i += gsz) {
        v4f l = c4[i];
        unsigned k0 = f2k(l.x), k1 = f2k(l.y), k2 = f2k(l.z), k3 = f2k(l.w);
        if ((k0 >> hshift) == prefix) atomicAdd(&lh[(k0 >> shift) & 0xFFu], 1u);
        if ((k1 >> hshift) == prefix) atomicAdd(&lh[(k1 >> shift) & 0xFFu], 1u);
        if ((k2 >> hshift) == prefix) atomicAdd(&lh[(k2 >> shift) & 0xFFu], 1u);
        if ((k3 >> hshift) == prefix) atomicAdd(&lh[(k3 >> shift) & 0xFFu], 1u);
    }
    for (int i = (n4 << 2) + gid; i < n; i += gsz) {
        unsigned k = f2k(cache[i]);
        if ((k >> hshift) == prefix) atomicAdd(&lh[(k >> shift) & 0xFFu], 1u);
    }
    __syncthreads();
    for (int i = tid; i < HIST_BINS; i += blockDim.x)
        if (lh[i]) atomicAdd(&hist[i], lh[i]);
}

// Fallback (no cache): recompute loss every pass (inputs re-read -> keep RT loads).
__global__ void tk_hist_nc(const float* __restrict__ pred, const float* __restrict__ targ,
                           const float* __restrict__ pwp, int n,
                           unsigned* __restrict__ hist, const unsigned* __restrict__ st,
                           int pass) {
    __shared__ unsigned lh[HIST_BINS];
    const int tid = threadIdx.x;
    for (int i = tid; i < HIST_BINS; i += blockDim.x) lh[i] = 0u;
    __syncthreads();

    const float    pw     = pwp[0];
    const unsigned prefix = st[0];
    const int shift = 24 - 8 * pass;
    const int gid = blockIdx.x * blockDim.x + tid;
    const int gsz = gridDim.x * blockDim.x;
    const int n4  = n >> 2;
    const v4f* p4 = (const v4f*)pred;
    const v4f* t4 = (const v4f*)targ;

    for (int i = gid; i < n4; i += gsz) {
        v4f p = p4[i], t = t4[i];
        float l[4] = { bce_loss(p.x, t.x, pw), bce_loss(p.y, t.y, pw),
                       bce_loss(p.z, t.z, pw), bce_loss(p.w, t.w, pw) };
        #pragma unroll
        for (int j = 0; j < 4; ++j) {
            unsigned k = f2k(l[j]);
            bool match = (pass == 0) || ((k >> (32 - 8 * pass)) == prefix);
            if (match) atomicAdd(&lh[(k >> shift) & 0xFFu], 1u);
        }
    }
    for (int i = (n4 << 2) + gid; i < n; i += gsz) {
        unsigned k = f2k(bce_loss(pred[i], targ[i], pw));
        bool match = (pass == 0) || ((k >> (32 - 8 * pass)) == prefix);
        if (match) atomicAdd(&lh[(k >> shift) & 0xFFu], 1u);
    }
    __syncthreads();
    for (int i = tid; i < HIST_BINS; i += blockDim.x)
        if (lh[i]) atomicAdd(&hist[i], lh[i]);
}

__global__ void tk_scan(unsigned* __restrict__ hist, unsigned* __restrict__ st, int pass) {
    unsigned prefix = st[0], krem = st[1], cgt = st[2];
    unsigned cum = 0u, sel = 0u;
    for (int d = HIST_BINS - 1; d >= 0; --d) {
        unsigned h = hist[d];
        if (cum + h >= krem) { sel = (unsigned)d; break; }
        cum += h;
    }
    cgt  += cum;
    krem -= cum;
    prefix = (prefix << 8) | sel;
    st[0] = prefix; st[1] = krem; st[2] = cgt;
    if (pass == 3) st[3] = prefix;
    for (int i = 0; i < HIST_BINS; ++i) hist[i] = 0u;
}

// ---- masked sum via V_WMMA_F32_16X16X4_F32 -------------------------------
// Wave tile = 128 cached losses (v4f/lane). A = 16x4 masked values,
// B = ones => C accumulates exact-f32 row sums; sum(C)/16 = wave total.
// Full tiles only (branch-free, EXEC all ones at every WMMA); tail -> tk_final.

__global__ void tk_sum_cache(const float* __restrict__ cache, int n,
                             const unsigned* __restrict__ st, float* __restrict__ partial) {
    const unsigned thresh = st[3];
    const int tid    = threadIdx.x;
    const int lane   = tid & 31;
    const int wave   = blockIdx.x * (blockDim.x >> 5) + (tid >> 5);
    const int nwaves = gridDim.x * (blockDim.x >> 5);
    const int ntiles = n >> 7;                       // full 128-elem tiles
    const v4f* c4 = (const v4f*)cache;

#if USE_WMMA_SUM
    v8f acc = {};
    v2f ones; ones[0] = 1.0f; ones[1] = 1.0f;
#else
    float acc_s = 0.0f;
#endif

    for (int t = wave; t < ntiles; t += nwaves) {
        v4f v = c4[t * 32 + lane];                   // global_load_b128, L2 hit
        float a0 = (f2k(v.x) > thresh) ? v.x : 0.0f;
        float a1 = (f2k(v.y) > thresh) ? v.y : 0.0f;
        float a2 = (f2k(v.z) > thresh) ? v.z : 0.0f;
        float a3 = (f2k(v.w) > thresh) ? v.w : 0.0f;
#if USE_WMMA_SUM
        v2f A0; A0[0] = a0; A0[1] = a1;
        v2f A1; A1[0] = a2; A1[1] = a3;
        acc = __builtin_amdgcn_wmma_f32_16x16x4_f32(false, A0, false, ones, (short)0, acc, false, false);
        acc = __builtin_amdgcn_wmma_f32_16x16x4_f32(false, A1, false, ones, (short)0, acc, false, false);
#else
        acc_s += (a0 + a1) + (a2 + a3);
#endif
    }

#if USE_WMMA_SUM
    float s = ((acc[0] + acc[1]) + (acc[2] + acc[3])) + ((acc[4] + acc[5]) + (acc[6] + acc[7]));
    s *= (1.0f / 16.0f);
#else
    float s = acc_s;
#endif

    __shared__ float red[NTHREADS];
    red[tid] = s;
    __syncthreads();
    for (int off = NTHREADS >> 1; off > 0; off >>= 1) {
        if (tid < off) red[tid] += red[tid + off];
        __syncthreads();
    }
    if (tid == 0) partial[blockIdx.x] = red[0];
}

__global__ void tk_sum_nc(const float* __restrict__ pred, const float* __restrict__ targ,
                          const float* __restrict__ pwp, int n,
                          const unsigned* __restrict__ st, float* __restrict__ partial) {
    const float    pw     = pwp[0];
    const unsigned thresh = st[3];
    const int tid    = threadIdx.x;
    const int lane   = tid & 31;
    const int wave   = blockIdx.x * (blockDim.x >> 5) + (tid >> 5);
    const int nwaves = gridDim.x * (blockDim.x >> 5);
    const int ntiles = n >> 6;                       // full 64-elem tiles
    const v2f* p2 = (const v2f*)pred;
    const v2f* t2 = (const v2f*)targ;

#if USE_WMMA_SUM
    v8f acc = {};
    v2f ones; ones[0] = 1.0f; ones[1] = 1.0f;
#else
    float acc_s = 0.0f;
#endif

    for (int t = wave; t < ntiles; t += nwaves) {
        v2f p  = p2[t * 32 + lane];
        v2f tt = t2[t * 32 + lane];
        float l0 = bce_loss(p.x, tt.x, pw);
        float l1 = bce_loss(p.y, tt.y, pw);
        float a0 = (f2k(l0) > thresh) ? l0 : 0.0f;
        float a1 = (f2k(l1) > thresh) ? l1 : 0.0f;
#if USE_WMMA_SUM
        v2f A; A[0] = a0; A[1] = a1;
        acc = __builtin_amdgcn_wmma_f32_16x16x4_f32(false, A, false, ones, (short)0, acc, false, false);
#else
        acc_s += a0 + a1;
#endif
    }

#if USE_WMMA_SUM
    float s = ((acc[0] + acc[1]) + (acc[2] + acc[3])) + ((acc[4] + acc[5]) + (acc[6] + acc[7]));
    s *= (1.0f / 16.0f);
#else
    float s = acc_s;
#endif

    __shared__ float red[NTHREADS];
    red[tid] = s;
    __syncthreads();
    for (int off = NTHREADS >> 1; off > 0; off >>= 1) {
        if (tid < off) red[tid] += red[tid + off];
        __syncthreads();
    }
    if (tid == 0) partial[blockIdx.x] = red[0];
}

// Final: fixed-order partial reduce + tail elements + tie handling.
__global__ void tk_final(const float* __restrict__ partial, int nb,
                         const unsigned* __restrict__ st, const int* __restrict__ kp, int n,
                         float* __restrict__ out,
                         const float* __restrict__ cache,
                         const float* __restrict__ pred, const float* __restrict__ targ,
                         const float* __restrict__ pwp,
                         int rem_start, int use_cache) {
    long long K = ((long long)n * (long long)kp[0]) / 100ll;
    if (K <= 0) { out[0] = 0.0f; return; }

    double s = 0.0;
    for (int i = 0; i < nb; ++i) s += (double)partial[i];   // fixed order

    const unsigned thresh = st[3];
    for (int i = rem_start; i < n; ++i) {                   // tail (< one wave tile)
        float l = use_cache ? cache[i] : bce_loss(pred[i], targ[i], pwp[0]);
        if (f2k(l) > thresh) s += (double)l;
    }

    float    T   = k2f(thresh);
    unsigned cgt = st[2];
    double mean = (s + (double)T * (double)(K - (long long)cgt)) / (double)K;
    out[0] = (float)mean;
}

// ---------- launcher ----------

extern "C" void kernel_launch(void* const* d_in, const int* in_sizes, int n_in,
                              void* d_out, int out_size, void* d_ws, size_t ws_size,
                              hipStream_t stream) {
    const float* pred = (const float*)d_in[0];
    const float* targ = (const float*)d_in[1];
    const float* pwp  = (const float*)d_in[2];
    const int*   kp   = (const int*)d_in[3];
    const int    n    = in_sizes[0];

    unsigned* hist    = (unsigned*)d_ws;                     // 256 u32
    unsigned* st      = hist + HIST_BINS;                    // state u32s
    float*    partial = (float*)(hist + 2 * HIST_BINS);      // NBLOCKS f32
    float*    cache   = (float*)((char*)d_ws + CACHE_OFF);   // n f32 (if room)

    const bool use_cache = ws_size >= (size_t)CACHE_OFF + (size_t)n * sizeof(float);

    tk_init<<<1, 1, 0, stream>>>(hist, st, kp, n);

    if (use_cache) {
        tk_p0_cache<<<NBLOCKS, NTHREADS, 0, stream>>>(pred, targ, pwp, n, hist, cache);
        tk_scan<<<1, 1, 0, stream>>>(hist, st, 0);
        for (int p = 1; p < 4; ++p) {
            tk_hist_cache<<<NBLOCKS, NTHREADS, 0, stream>>>(cache, n, hist, st, p);
            tk_scan<<<1, 1, 0, stream>>>(hist, st, p);
        }
        tk_sum_cache<<<NBLOCKS, NTHREADS, 0, stream>>>(cache, n, st, partial);
        tk_final<<<1, 1, 0, stream>>>(partial, NBLOCKS, st, kp, n, (float*)d_out,
                                      cache, nullptr, nullptr, nullptr,
                                      (n >> 7) << 7, 1);
    } else {
        for (int p = 0; p < 4; ++p) {
            tk_hist_nc<<<NBLOCKS, NTHREADS, 0, stream>>>(pred, targ, pwp, n, hist, st, p);
            tk_scan<<<1, 1, 0, stream>>>(hist, st, p);
        }
        tk_sum_nc<<<NBLOCKS, NTHREADS, 0, stream>>>(pred, targ, pwp, n, st, partial);
        tk_final<<<1, 1, 0, stream>>>(partial, NBLOCKS, st, kp, n, (float*)d_out,
                                      nullptr, pred, targ, pwp,
                                      (n >> 6) << 6, 0);
    }
}